// MULTIGCN_16810501996622
// MI455X (gfx1250) — compile-verified
//
#include <hip/hip_runtime.h>
#include <hip/hip_bf16.h>

typedef float v2f __attribute__((ext_vector_type(2)));
typedef float v8f __attribute__((ext_vector_type(8)));

#define ATOMIC_FADD(p, v) __hip_atomic_fetch_add((p), (v), __ATOMIC_RELAXED, __HIP_MEMORY_SCOPE_AGENT)

// ---------------------------------------------------------------------------
// Copy x [N,128] into h_cat cols 0..127 (row stride 512). float4 granularity.
// ---------------------------------------------------------------------------
__global__ __launch_bounds__(256) void copy_x_kernel(const float* __restrict__ x,
                                                     float* __restrict__ hcat,
                                                     int n) {
    int gid = blockIdx.x * 256 + threadIdx.x;     // over n*32 float4s
    if (gid >= n * 32) return;
    int i = gid >> 5;
    int q = gid & 31;
    float4 v = ((const float4*)x)[(size_t)i * 32 + q];
    ((float4*)(hcat + (size_t)i * 512))[q] = v;
}

// ---------------------------------------------------------------------------
// Degree init (self loop contributes 1), accumulate, rsqrt in place.
// ---------------------------------------------------------------------------
__global__ __launch_bounds__(256) void deg_init_kernel(float* __restrict__ deg, int n) {
    int i = blockIdx.x * 256 + threadIdx.x;
    if (i < n) deg[i] = 1.0f;
}

__global__ __launch_bounds__(256) void deg_accum_kernel(const int* __restrict__ dst,
                                                        float* __restrict__ deg, int e) {
    int i = blockIdx.x * 256 + threadIdx.x;
    if (i < e) ATOMIC_FADD(&deg[dst[i]], 1.0f);
}

__global__ __launch_bounds__(256) void deg_rsqrt_kernel(float* __restrict__ deg, int n) {
    int i = blockIdx.x * 256 + threadIdx.x;
    if (i < n) deg[i] = rsqrtf(deg[i]);          // deg >= 1 always (self loops)
}

// norm[e] = dis[src]*dis[dst], computed once, reused by all 3 layers
__global__ __launch_bounds__(256) void norm_kernel(const int* __restrict__ src,
                                                   const int* __restrict__ dst,
                                                   const float* __restrict__ dis,
                                                   float* __restrict__ norm, int e) {
    int i = blockIdx.x * 256 + threadIdx.x;
    if (i < e) norm[i] = dis[src[i]] * dis[dst[i]];
}

// ---------------------------------------------------------------------------
// WMMA f32 GEMM with fused self-loop + bias epilogue:
//   h_tmp[i][c] = (A @ W)[i][c]                       (dense row store, for gathers)
//   hout [i][c] = h_tmp[i][c]*dis[i]^2 + b[c]         (into concat buffer, stride 512)
// Block = 256 threads = 8 waves; tile = 16 rows x 128 cols; wave owns 16x16.
// ---------------------------------------------------------------------------
__global__ __launch_bounds__(256) void gemm128_wmma_kernel(const float* __restrict__ A,
                                                           int lda,
                                                           const float* __restrict__ W,
                                                           const float* __restrict__ dis,
                                                           const float* __restrict__ bias,
                                                           float* __restrict__ h_tmp,
                                                           float* __restrict__ hout,
                                                           int nrows) {
    __shared__ float As[16 * 132];
    const int t    = threadIdx.x;
    const int m0   = blockIdx.x * 16;
    const int lane = t & 31;
    const int wave = t >> 5;
    const int n0   = wave * 16;
    const int half = lane >> 4;     // 0: lanes 0-15, 1: lanes 16-31
    const int ml   = lane & 15;
    const bool full = (m0 + 16 <= nrows);

    // Stage 16x128 A tile (coalesced); tail tile zero-padded.
    if (full) {
        for (int i = t; i < 16 * 128; i += 256) {
            int r = i >> 7, c = i & 127;
            As[r * 132 + c] = A[(size_t)(m0 + r) * lda + c];
        }
    } else {
        for (int i = t; i < 16 * 128; i += 256) {
            int r = i >> 7, c = i & 127;
            As[r * 132 + c] = (m0 + r < nrows) ? A[(size_t)(m0 + r) * lda + c] : 0.0f;
        }
    }
    __syncthreads();

    v8f acc = {};
    #pragma unroll 4
    for (int k = 0; k < 128; k += 4) {
        const int kk = k + 2 * half;
        v2f a;
        a.x = As[ml * 132 + kk];
        a.y = As[ml * 132 + kk + 1];
        v2f b;
        const float* Bp = W + (size_t)kk * 128 + n0 + ml;
        b.x = Bp[0];
        b.y = Bp[128];
        acc = __builtin_amdgcn_wmma_f32_16x16x4_f32(false, a, false, b,
                                                    (short)0, acc, false, false);
    }

    // D layout: VGPR r -> row (r + 8*half), col = ml
    const float bb = bias[n0 + ml];
    if (full) {
        #pragma unroll
        for (int r = 0; r < 8; ++r) {
            int row  = m0 + r + half * 8;
            float dv = dis[row];               // broadcast within half-wave
            float v  = acc[r];
            h_tmp[(size_t)row * 128 + n0 + ml] = v;
            hout [(size_t)row * 512 + n0 + ml] = v * dv * dv + bb;
        }
    } else {
        #pragma unroll
        for (int r = 0; r < 8; ++r) {
            int row = m0 + r + half * 8;
            if (row < nrows) {
                float dv = dis[row];
                float v  = acc[r];
                h_tmp[(size_t)row * 128 + n0 + ml] = v;
                hout [(size_t)row * 512 + n0 + ml] = v * dv * dv + bb;
            }
        }
    }
}

// ---------------------------------------------------------------------------
// Edge scatter: one wave (32 lanes) per edge; lane handles 4 channels.
// hout[dst] += h[src] * norm[e]  via hardware global_atomic_add_f32 (L2-resident).
// ---------------------------------------------------------------------------
__global__ __launch_bounds__(256) void edge_scatter_kernel(const float* __restrict__ h,
                                                           const float* __restrict__ norm,
                                                           const int* __restrict__ src,
                                                           const int* __restrict__ dst,
                                                           float* __restrict__ hout,
                                                           int e) {
    int gid  = blockIdx.x * 256 + threadIdx.x;   // over e*32
    int edge = gid >> 5;
    int lane = gid & 31;
    if (edge >= e) return;
    int s   = src[edge];                          // broadcast loads
    int d   = dst[edge];
    float w = norm[edge];
    float4 v = ((const float4*)(h + (size_t)s * 128))[lane];  // coalesced 512B/edge
    float* o = hout + (size_t)d * 512 + (size_t)lane * 4;
    v.x *= w; v.y *= w; v.z *= w; v.w *= w;
    asm volatile(
        "global_atomic_add_f32 %0, %1, off\n\t"
        "global_atomic_add_f32 %0, %2, off offset:4\n\t"
        "global_atomic_add_f32 %0, %3, off offset:8\n\t"
        "global_atomic_add_f32 %0, %4, off offset:12"
        :
        : "v"(o), "v"(v.x), "v"(v.y), "v"(v.z), "v"(v.w)
        : "memory");
    // no explicit storecnt wait needed: S_ENDPGM performs an implicit wait-idle
}

// ---------------------------------------------------------------------------
// Final head: pred[N,16] = hcat[N,512] @ W_lin[512,16] + b_lin.
// One wave per 16-row tile; single 16x16 WMMA accumulator, K=512.
// ---------------------------------------------------------------------------
__global__ __launch_bounds__(32) void final_linear_wmma_kernel(const float* __restrict__ hcat,
                                                               const float* __restrict__ Wlin,
                                                               const float* __restrict__ blin,
                                                               float* __restrict__ pred,
                                                               int nrows) {
    const int m0   = blockIdx.x * 16;
    const int lane = threadIdx.x & 31;
    const int half = lane >> 4;
    const int ml   = lane & 15;
    const bool full = (m0 + 16 <= nrows);

    int arow = m0 + ml;
    if (arow >= nrows) arow = nrows - 1;          // clamp (tail tile only)
    const float* Ap = hcat + (size_t)arow * 512;

    v8f acc = {};
    #pragma unroll 4
    for (int k = 0; k < 512; k += 4) {
        const int kk = k + 2 * half;
        v2f a;
        a.x = Ap[kk];
        a.y = Ap[kk + 1];
        v2f b;
        const float* Bp = Wlin + (size_t)kk * 16 + ml;
        b.x = Bp[0];
        b.y = Bp[16];
        acc = __builtin_amdgcn_wmma_f32_16x16x4_f32(false, a, false, b,
                                                    (short)0, acc, false, false);
    }
    float bb = blin[ml];
    if (full) {
        #pragma unroll
        for (int r = 0; r < 8; ++r) {
            int row = m0 + r + half * 8;
            pred[(size_t)row * 16 + ml] = acc[r] + bb;
        }
    } else {
        #pragma unroll
        for (int r = 0; r < 8; ++r) {
            int row = m0 + r + half * 8;
            if (row < nrows) pred[(size_t)row * 16 + ml] = acc[r] + bb;
        }
    }
}

// ---------------------------------------------------------------------------
extern "C" void kernel_launch(void* const* d_in, const int* in_sizes, int n_in,
                              void* d_out, int out_size, void* d_ws, size_t ws_size,
                              hipStream_t stream) {
    const float* x    = (const float*)d_in[0];
    const int*   ei   = (const int*)d_in[1];          // [2,E] int32
    const float* W[3] = {(const float*)d_in[2], (const float*)d_in[4], (const float*)d_in[6]};
    const float* B[3] = {(const float*)d_in[3], (const float*)d_in[5], (const float*)d_in[7]};
    const float* Wlin = (const float*)d_in[8];
    const float* blin = (const float*)d_in[9];

    const int N = in_sizes[0] / 128;
    const int E = in_sizes[1] / 2;
    const int* src = ei;
    const int* dst = ei + E;

    float* pred = (float*)d_out;                       // [N,16]
    float* hcat = (float*)d_out + (size_t)N * 16;      // [N,512]

    // workspace layout: dis [N] | norm [E] | h_tmp [N*128]   (256B aligned)
    char* wsp = (char*)d_ws;
    float* dis  = (float*)wsp;
    size_t off  = (((size_t)N * 4) + 255) & ~(size_t)255;
    float* norm = (float*)(wsp + off);
    off += (((size_t)E * 4) + 255) & ~(size_t)255;
    float* h_tmp = (float*)(wsp + off);

    const int tilesN = (N + 15) / 16;

    // x -> concat cols 0..127
    copy_x_kernel<<<(N * 32 + 255) / 256, 256, 0, stream>>>(x, hcat, N);

    // dis = rsqrt(1 + in-degree); norm per edge
    deg_init_kernel<<<(N + 255) / 256, 256, 0, stream>>>(dis, N);
    deg_accum_kernel<<<(E + 255) / 256, 256, 0, stream>>>(dst, dis, E);
    deg_rsqrt_kernel<<<(N + 255) / 256, 256, 0, stream>>>(dis, N);
    norm_kernel<<<(E + 255) / 256, 256, 0, stream>>>(src, dst, dis, norm, E);

    // three GCN layers; layer l reads concat cols (l-1)*128, writes cols l*128
    for (int l = 1; l <= 3; ++l) {
        const float* Ain  = hcat + (size_t)(l - 1) * 128;   // lda = 512
        float*       hout = hcat + (size_t)l * 128;         // stride 512
        gemm128_wmma_kernel<<<tilesN, 256, 0, stream>>>(Ain, 512, W[l - 1], dis, B[l - 1],
                                                        h_tmp, hout, N);
        edge_scatter_kernel<<<(int)(((size_t)E * 32 + 255) / 256), 256, 0, stream>>>(
            h_tmp, norm, src, dst, hout, E);
    }

    // classification head
    final_linear_wmma_kernel<<<tilesN, 32, 0, stream>>>(hcat, Wlin, blin, pred, N);
}